// mLSTMcell_45956150067798
// MI455X (gfx1250) — compile-verified
//
#include <hip/hip_runtime.h>
#include <hip/hip_bf16.h>
#include <math.h>

typedef __attribute__((ext_vector_type(2))) float v2f;
typedef __attribute__((ext_vector_type(4))) float f4;
typedef __attribute__((ext_vector_type(8))) float v8f;

#define BATCH 128
#define DH    1024   // hidden = input = 1024
#define BD    (BATCH * DH)

// ---------------------------------------------------------------------------
// Kernel 1: q/k/v/o projections via fp32 WMMA (V_WMMA_F32_16X16X4_F32).
// One wave (32 threads) computes one 16x16 tile of out = x @ W (+bias, postop).
// grid = (DH/16 n-tiles, BATCH/16 m-tiles, 4 matrices), block = 32.
//
// A (16x4 f32) per ISA 7.12.2: lane L<16 holds M=L, K={0,1}; lane L+16 holds
// M=L, K={2,3}.  B (4x16 f32): V0 = rows K0 (lanes 0-15) / K2 (lanes 16-31),
// V1 = rows K1 / K3, N = lane&15.  C/D: vgpr r holds M=r(+8), N=lane&15.
// ---------------------------------------------------------------------------
__global__ void __launch_bounds__(32)
proj_wmma_kernel(const float* __restrict__ x,
                 const float* __restrict__ Wq, const float* __restrict__ bq,
                 const float* __restrict__ Wk, const float* __restrict__ bk,
                 const float* __restrict__ Wv, const float* __restrict__ bv,
                 const float* __restrict__ Wo, const float* __restrict__ bo,
                 float* __restrict__ q, float* __restrict__ k,
                 float* __restrict__ v, float* __restrict__ o) {
    const int lane = threadIdx.x;          // 0..31 (wave32)
    const int n0   = blockIdx.x * 16;
    const int m0   = blockIdx.y * 16;
    const int mat  = blockIdx.z;           // 0=q 1=k 2=v 3=o

    const float* W;  const float* bias;  float* out;
    if (mat == 0)      { W = Wq; bias = bq; out = q; }
    else if (mat == 1) { W = Wk; bias = bk; out = k; }
    else if (mat == 2) { W = Wv; bias = bv; out = v; }
    else               { W = Wo; bias = bo; out = o; }

    const int nh = lane & 15;              // M for A-frag, N for B-frag
    const int kh = (lane >> 4) * 2;        // 0 or 2

    const float* arow = x + (size_t)(m0 + nh) * DH;   // A row (M = nh)
    const float* bcol = W + (size_t)(n0 + nh);        // B column (N = nh)

    v8f acc = {};
#pragma unroll 4
    for (int k0 = 0; k0 < DH; k0 += 4) {
        v2f a, b;
        a.x = arow[k0 + kh];
        a.y = arow[k0 + kh + 1];
        b.x = bcol[(size_t)(k0 + kh)     * DH];
        b.y = bcol[(size_t)(k0 + kh + 1) * DH];
        acc = __builtin_amdgcn_wmma_f32_16x16x4_f32(
            /*neg_a=*/false, a, /*neg_b=*/false, b,
            /*c_mod=*/(short)0, acc, /*reuse_a=*/false, /*reuse_b=*/false);
    }

    // Epilogue: bias + per-matrix postop, scatter per C/D layout.
    const float bb    = bias[n0 + nh];
    const int   mbase = m0 + (lane >> 4) * 8;
#pragma unroll
    for (int r = 0; r < 8; ++r) {
        float val = acc[r] + bb;
        if (mat == 1)      val *= 0.03125f;                   // k * 1/sqrt(1024)
        else if (mat == 3) val = 1.0f / (1.0f + expf(-val));  // sigmoid(o)
        out[(size_t)(mbase + r) * DH + (n0 + nh)] = val;
    }
}

// ---------------------------------------------------------------------------
// Kernel 2: per-batch scalar gates + normalizer state + denominator.
//   i = exp(x.w_i + b_i), f = sigmoid(x.w_f + b_f)
//   n_t = f*n_prev + i*k            (written to d_out's n_t slab)
//   den = max(|n_t . q|, 1)
// grid = BATCH, block = 256.
// ---------------------------------------------------------------------------
__global__ void __launch_bounds__(256)
gates_norm_kernel(const float* __restrict__ x, const float* __restrict__ nprev,
                  const float* __restrict__ wi, const float* __restrict__ bi,
                  const float* __restrict__ wf, const float* __restrict__ bf,
                  const float* __restrict__ q,  const float* __restrict__ k,
                  float* __restrict__ ig, float* __restrict__ fg,
                  float* __restrict__ den, float* __restrict__ nout) {
    __shared__ float s1[256], s2[256];
    __shared__ float sIv, sFv;
    const int b = blockIdx.x, tid = threadIdx.x;
    const float* xr = x + (size_t)b * DH;

    float si = 0.f, sf = 0.f;
    for (int e = tid; e < DH; e += 256) {
        const float xv = xr[e];
        si = fmaf(xv, wi[e], si);
        sf = fmaf(xv, wf[e], sf);
    }
    s1[tid] = si; s2[tid] = sf; __syncthreads();
    for (int off = 128; off > 0; off >>= 1) {
        if (tid < off) { s1[tid] += s1[tid + off]; s2[tid] += s2[tid + off]; }
        __syncthreads();
    }
    if (tid == 0) {
        const float iv = expf(s1[0] + bi[0]);
        const float fv = 1.0f / (1.0f + expf(-(s2[0] + bf[0])));
        ig[b] = iv; fg[b] = fv; sIv = iv; sFv = fv;
    }
    __syncthreads();
    const float iv = sIv, fv = sFv;

    const float* kr = k     + (size_t)b * DH;
    const float* qr = q     + (size_t)b * DH;
    const float* nr = nprev + (size_t)b * DH;
    float acc = 0.f;
    for (int e = tid; e < DH; e += 256) {
        const float nt = fmaf(fv, nr[e], iv * kr[e]);
        nout[(size_t)b * DH + e] = nt;
        acc = fmaf(nt, qr[e], acc);
    }
    s1[tid] = acc; __syncthreads();
    for (int off = 128; off > 0; off >>= 1) {
        if (tid < off) s1[tid] += s1[tid + off];
        __syncthreads();
    }
    if (tid == 0) den[b] = fmaxf(fabsf(s1[0]), 1.0f);
}

// ---------------------------------------------------------------------------
// Kernel 3: the 1 GB memory stream (dominant cost, ~46 us roofline).
//   C_t[b,d,e] = f*C_prev[b,d,e] + (i*v[b,d])*k[b,e]
//   num[b,d]   = sum_e C_t[b,d,e]*q[b,e]    (fused while data is in regs)
//   h[b,d]     = o[b,d]*num/den[b]
// One wave per (b,d) row; q/k rows staged in LDS; C traffic via nontemporal
// float4 so it does not evict the hot per-batch vectors from L2.
// grid = (DH/8, BATCH), block = 256 (8 waves).
// ---------------------------------------------------------------------------
__global__ void __launch_bounds__(256)
cell_stream_kernel(const float* __restrict__ Cprev,
                   const float* __restrict__ q, const float* __restrict__ k,
                   const float* __restrict__ v, const float* __restrict__ o,
                   const float* __restrict__ ig, const float* __restrict__ fg,
                   const float* __restrict__ den,
                   float* __restrict__ h, float* __restrict__ Cout) {
    __shared__ float qs[DH], ks[DH];
    const int b   = blockIdx.y;
    const int tid = threadIdx.x;

    // Cooperative 16B-wide stage of q/k rows (1024 floats each).
    ((f4*)qs)[tid] = ((const f4*)(q + (size_t)b * DH))[tid];
    ((f4*)ks)[tid] = ((const f4*)(k + (size_t)b * DH))[tid];
    __syncthreads();

    const int wave = tid >> 5, lane = tid & 31;
    const int d = blockIdx.x * 8 + wave;

    const float  fv     = fg[b];
    const float  iv     = ig[b] * v[(size_t)b * DH + d];
    const size_t rowoff = ((size_t)b * DH + d) * DH;

    float acc = 0.f;
#pragma unroll
    for (int it = 0; it < 8; ++it) {
        const int e = it * 128 + lane * 4;
        f4 cp = __builtin_nontemporal_load((const f4*)(Cprev + rowoff + e));
        f4 kk = *(const f4*)(ks + e);
        f4 qq = *(const f4*)(qs + e);
        f4 ct = fv * cp + iv * kk;
        __builtin_nontemporal_store(ct, (f4*)(Cout + rowoff + e));
        f4 pr = ct * qq;
        acc += pr.x + pr.y + pr.z + pr.w;
    }
    // wave32 reduction of the row dot-product
    for (int off = 16; off > 0; off >>= 1)
        acc += __shfl_xor(acc, off, 32);
    if (lane == 0)
        h[(size_t)b * DH + d] = o[(size_t)b * DH + d] * acc / den[b];
}

// ---------------------------------------------------------------------------
extern "C" void kernel_launch(void* const* d_in, const int* in_sizes, int n_in,
                              void* d_out, int out_size, void* d_ws, size_t ws_size,
                              hipStream_t stream) {
    const float* x     = (const float*)d_in[0];
    const float* Cprev = (const float*)d_in[1];
    const float* nprev = (const float*)d_in[2];
    const float* Wq = (const float*)d_in[3];  const float* bq = (const float*)d_in[4];
    const float* Wk = (const float*)d_in[5];  const float* bk = (const float*)d_in[6];
    const float* Wv = (const float*)d_in[7];  const float* bv = (const float*)d_in[8];
    const float* Wo = (const float*)d_in[9];  const float* bo = (const float*)d_in[10];
    const float* wi = (const float*)d_in[11]; const float* bi = (const float*)d_in[12];
    const float* wf = (const float*)d_in[13]; const float* bf = (const float*)d_in[14];

    float* ws  = (float*)d_ws;
    float* q   = ws;
    float* k   = ws + (size_t)BD;
    float* v   = ws + (size_t)2 * BD;
    float* o   = ws + (size_t)3 * BD;
    float* ig  = ws + (size_t)4 * BD;
    float* fg  = ig + BATCH;
    float* den = fg + BATCH;

    float* out  = (float*)d_out;
    float* h    = out;                                          // [B, DH]
    float* Cout = out + (size_t)BD;                             // [B, DH, DH]
    float* nout = out + (size_t)BD + (size_t)BATCH * DH * DH;   // [B, DH]

    // 1) projections (WMMA fp32)
    proj_wmma_kernel<<<dim3(DH / 16, BATCH / 16, 4), 32, 0, stream>>>(
        x, Wq, bq, Wk, bk, Wv, bv, Wo, bo, q, k, v, o);

    // 2) scalar gates, n_t, den
    gates_norm_kernel<<<BATCH, 256, 0, stream>>>(
        x, nprev, wi, bi, wf, bf, q, k, ig, fg, den, nout);

    // 3) C stream + fused einsum + h
    cell_stream_kernel<<<dim3(DH / 8, BATCH), 256, 0, stream>>>(
        Cprev, q, k, v, o, ig, fg, den, h, Cout);
}